// GraphSAGE_21096879358044
// MI455X (gfx1250) — compile-verified
//
#include <hip/hip_runtime.h>
#include <hip/hip_bf16.h>

// GraphSAGE 2-layer forward for MI455X (gfx1250, wave32).
// Scatter phase is L2-resident (total working set ~80MB < 192MB L2), so it is
// atomic/L2-bandwidth bound: coalesced float4 gathers + global_atomic_add_f32.
// Layer-1 GEMM (50000x128x256 fp32) uses V_WMMA_F32_16X16X4_F32 with W staged
// in LDS; layer-2 (N=2) is a wave-per-node dot product with shuffle reduction.

typedef float v2f __attribute__((ext_vector_type(2)));
typedef float v8f __attribute__((ext_vector_type(8)));

#define D_FEAT 128

// ---------------------------------------------------------------- zero scratch
__global__ void zero_kernel(float* __restrict__ p, long long n) {
    long long i = (long long)blockIdx.x * blockDim.x + threadIdx.x;
    long long stride = (long long)gridDim.x * blockDim.x;
    for (; i < n; i += stride) p[i] = 0.0f;
}

// ------------------------------------------------------- edge scatter (mean num)
// One wave per edge: 32 lanes x float4 = the full 512B feature row.
__global__ void scatter_kernel(const float* __restrict__ feat,
                               const long long* __restrict__ ei,
                               float* __restrict__ agg,
                               float* __restrict__ cnt,
                               int nEdges, int doCount) {
    int wave = (blockIdx.x * blockDim.x + threadIdx.x) >> 5;
    int lane = threadIdx.x & 31;
    if (wave >= nEdges) return;
    long long src = ei[wave];
    long long dst = ei[(long long)nEdges + wave];
    const float4* row = (const float4*)(feat + src * D_FEAT);
    float4 v = row[lane];
    float* base = agg + dst * D_FEAT + lane * 4;
    atomicAdd(base + 0, v.x);
    atomicAdd(base + 1, v.y);
    atomicAdd(base + 2, v.z);
    atomicAdd(base + 3, v.w);
    if (doCount && lane == 0) atomicAdd(&cnt[dst], 1.0f);
}

// ------------------------------------------------- layer-1 fused GEMM (WMMA f32)
// h = relu(mean @ W1l + b1l + x @ W1r), mean = agg * (1/max(cnt,1))
// Block: 256 threads = 8 waves; each wave computes a 16x128 output strip.
// K loop: 32 steps of V_WMMA_F32_16X16X4_F32 per phase (phase0: mean/W1l,
// phase1: x/W1r), accumulating into the same 8 v8f accumulators.
__global__ __launch_bounds__(256)
void sage1_wmma_kernel(const float* __restrict__ x,
                       const float* __restrict__ agg,
                       const float* __restrict__ cnt,
                       const float* __restrict__ W1l,
                       const float* __restrict__ b1l,
                       const float* __restrict__ W1r,
                       float* __restrict__ h,
                       int nNodes) {
    __shared__ float sW[D_FEAT * D_FEAT];   // 64 KB: W1l then W1r

    const int tid   = threadIdx.x;
    const int wv    = tid >> 5;             // wave in block, 0..7
    const int lane  = tid & 31;
    const int half  = lane >> 4;            // 0: K%4 in {0,1}; 1: K%4 in {2,3}
    const int nlane = lane & 15;            // column within 16-wide tile / row M
    const int rowBase = blockIdx.x * 128 + wv * 16;

    // A-fragment row for this lane (clamped; stores are guarded later)
    int aRow = rowBase + nlane;
    if (aRow >= nNodes) aRow = nNodes - 1;
    const float inv = 1.0f / fmaxf(cnt[aRow], 1.0f);

    v8f acc[8];
#pragma unroll
    for (int n = 0; n < 8; ++n) acc[n] = (v8f){0.f,0.f,0.f,0.f,0.f,0.f,0.f,0.f};

    // -------- phase 0: stage W1l, accumulate mean @ W1l
    {
        const float4* wsrc = (const float4*)W1l;
        float4* wdst = (float4*)sW;
        for (int i = tid; i < (D_FEAT * D_FEAT) / 4; i += 256) wdst[i] = wsrc[i];
    }
    __syncthreads();
    {
        const float* arow = agg + (long long)aRow * D_FEAT;
#pragma unroll 4
        for (int kk = 0; kk < 32; ++kk) {
            const int k0 = 4 * kk + 2 * half;
            v2f a;
            a.x = arow[k0 + 0] * inv;
            a.y = arow[k0 + 1] * inv;
#pragma unroll
            for (int n = 0; n < 8; ++n) {
                v2f b;
                b.x = sW[(k0 + 0) * D_FEAT + 16 * n + nlane];
                b.y = sW[(k0 + 1) * D_FEAT + 16 * n + nlane];
                acc[n] = __builtin_amdgcn_wmma_f32_16x16x4_f32(
                    false, a, false, b, (short)0, acc[n], false, false);
            }
        }
    }
    __syncthreads();

    // -------- phase 1: stage W1r, accumulate x @ W1r into same accumulators
    {
        const float4* wsrc = (const float4*)W1r;
        float4* wdst = (float4*)sW;
        for (int i = tid; i < (D_FEAT * D_FEAT) / 4; i += 256) wdst[i] = wsrc[i];
    }
    __syncthreads();
    {
        const float* arow = x + (long long)aRow * D_FEAT;
#pragma unroll 4
        for (int kk = 0; kk < 32; ++kk) {
            const int k0 = 4 * kk + 2 * half;
            v2f a;
            a.x = arow[k0 + 0];
            a.y = arow[k0 + 1];
#pragma unroll
            for (int n = 0; n < 8; ++n) {
                v2f b;
                b.x = sW[(k0 + 0) * D_FEAT + 16 * n + nlane];
                b.y = sW[(k0 + 1) * D_FEAT + 16 * n + nlane];
                acc[n] = __builtin_amdgcn_wmma_f32_16x16x4_f32(
                    false, a, false, b, (short)0, acc[n], false, false);
            }
        }
    }

    // -------- epilogue: bias + relu, guarded store
#pragma unroll
    for (int n = 0; n < 8; ++n) {
        const int col = 16 * n + nlane;
        const float bias = b1l[col];
#pragma unroll
        for (int r = 0; r < 8; ++r) {
            const int m = rowBase + r + 8 * half;   // C/D layout: M = r + 8*half
            if (m < nNodes) {
                float v = acc[n][r] + bias;
                h[(long long)m * D_FEAT + col] = fmaxf(v, 0.0f);
            }
        }
    }
}

// --------------------------------------------- layer-2 (N=2): wave per node
__global__ __launch_bounds__(256)
void sage2_kernel(const float* __restrict__ agg,
                  const float* __restrict__ cnt,
                  const float* __restrict__ h,
                  const float* __restrict__ W2l,
                  const float* __restrict__ b2l,
                  const float* __restrict__ W2r,
                  float* __restrict__ out,
                  int nNodes) {
    int node = (blockIdx.x * blockDim.x + threadIdx.x) >> 5;
    int lane = threadIdx.x & 31;
    if (node >= nNodes) return;
    const float inv = 1.0f / fmaxf(cnt[node], 1.0f);
    const float* aRow = agg + (long long)node * D_FEAT;
    const float* hRow = h + (long long)node * D_FEAT;
    float s0 = 0.f, s1 = 0.f;
#pragma unroll
    for (int j = 0; j < 4; ++j) {
        int k = lane + 32 * j;
        float m  = aRow[k] * inv;
        float hv = hRow[k];
        s0 += m * W2l[k * 2 + 0] + hv * W2r[k * 2 + 0];
        s1 += m * W2l[k * 2 + 1] + hv * W2r[k * 2 + 1];
    }
#pragma unroll
    for (int off = 16; off >= 1; off >>= 1) {
        s0 += __shfl_xor(s0, off, 32);
        s1 += __shfl_xor(s1, off, 32);
    }
    if (lane == 0) {
        out[(long long)node * 2 + 0] = s0 + b2l[0];
        out[(long long)node * 2 + 1] = s1 + b2l[1];
    }
}

// ------------------------------------------------------------------- launcher
extern "C" void kernel_launch(void* const* d_in, const int* in_sizes, int n_in,
                              void* d_out, int out_size, void* d_ws, size_t ws_size,
                              hipStream_t stream) {
    const float*     x   = (const float*)d_in[0];
    const long long* ei  = (const long long*)d_in[1];   // [2, E] int64
    const float*     W1l = (const float*)d_in[2];
    const float*     b1l = (const float*)d_in[3];
    const float*     W1r = (const float*)d_in[4];
    const float*     W2l = (const float*)d_in[5];
    const float*     b2l = (const float*)d_in[6];
    const float*     W2r = (const float*)d_in[7];
    float*           out = (float*)d_out;

    const int nNodes = in_sizes[0] / D_FEAT;            // 50000
    const int nEdges = in_sizes[1] / 2;                 // 800000
    const long long NF = (long long)nNodes * D_FEAT;    // 6.4M floats

    // workspace layout (floats): cnt | agg | h
    float* ws  = (float*)d_ws;
    float* cnt = ws;
    float* agg = ws + 51200;                            // cnt padded
    float* h   = agg + NF;

    const int ZB = 1024;
    const int scatterBlocks = (nEdges + 7) / 8;         // 8 waves per 256-thr block
    const int gemmBlocks    = (nNodes + 127) / 128;     // 128 rows per block
    const int l2Blocks      = (nNodes + 7) / 8;         // 8 waves per block

    // Layer 1
    zero_kernel<<<2048, ZB, 0, stream>>>(ws, 51200 + NF);               // cnt+agg
    scatter_kernel<<<scatterBlocks, 256, 0, stream>>>(x, ei, agg, cnt, nEdges, 1);
    sage1_wmma_kernel<<<gemmBlocks, 256, 0, stream>>>(x, agg, cnt, W1l, b1l, W1r,
                                                      h, nNodes);
    // Layer 2 (reuse cnt; agg re-zeroed)
    zero_kernel<<<2048, ZB, 0, stream>>>(agg, NF);
    scatter_kernel<<<scatterBlocks, 256, 0, stream>>>(h, ei, agg, cnt, nEdges, 0);
    sage2_kernel<<<l2Blocks, 256, 0, stream>>>(agg, cnt, h, W2l, b2l, W2r,
                                               out, nNodes);
}